// ChebAugmentedLayer_21303037788635
// MI455X (gfx1250) — compile-verified
//
#include <hip/hip_runtime.h>
#include <stdint.h>

#define N_NODES 50000
#define N_EDGES 800000
#define FDIM 128
#define NEIG 64
#define HDIM 128
#define BN_EPS 1e-5f
#define LDS_ROW 132   // 128 data dwords + 4 pad dwords (TDM pad) -> conflict-free banking

typedef float v2f __attribute__((ext_vector_type(2)));
typedef float v8f __attribute__((ext_vector_type(8)));
typedef unsigned int u32x4 __attribute__((ext_vector_type(4)));
typedef int i32x8 __attribute__((ext_vector_type(8)));
typedef int i32x4 __attribute__((ext_vector_type(4)));

// ---------------------------------------------------------------- utilities
__global__ void zero_f32(float* p, long n) {
  long i = (long)blockIdx.x * blockDim.x + threadIdx.x;
  long stride = (long)gridDim.x * blockDim.x;
  for (; i < n; i += stride) p[i] = 0.0f;
}

// ---------------------------------------------------------------- degree
__global__ void deg_kernel(const long long* __restrict__ dst, float* __restrict__ deg) {
  int e = blockIdx.x * blockDim.x + threadIdx.x;
  if (e < N_EDGES) atomicAdd(&deg[(int)dst[e]], 1.0f);
}

__global__ void dsqrt_kernel(const float* __restrict__ deg, float* __restrict__ dsq) {
  int i = blockIdx.x * blockDim.x + threadIdx.x;
  if (i < N_NODES) {
    float d = deg[i];
    d = d < 1.0f ? 1.0f : d;
    dsq[i] = rsqrtf(d);
  }
}

// ---------------------------------------------------------------- SpMM scatter
// one block (128 threads = 128 features) per edge; agg[dst] += x[src]*dsq[src]
__global__ void edge_scatter(const float* __restrict__ x, const float* __restrict__ dsq,
                             const long long* __restrict__ src, const long long* __restrict__ dst,
                             float* __restrict__ agg) {
  int e = blockIdx.x;
  int f = threadIdx.x;
  int s = (int)src[e];
  int d = (int)dst[e];
  atomicAdd(&agg[(long)d * FDIM + f], x[(long)s * FDIM + f] * dsq[s]);
}

// Xi from Laplacian result h (needs *dsq[dst]) and previous blocks
__global__ void cheb_combine(float* __restrict__ Xo, const float* __restrict__ h,
                             const float* __restrict__ dsq, const float* __restrict__ Xp,
                             const float* __restrict__ Xpp, const float* __restrict__ lmax,
                             int first) {
  long i = (long)blockIdx.x * blockDim.x + threadIdx.x;
  const long n = (long)N_NODES * FDIM;
  long stride = (long)gridDim.x * blockDim.x;
  float rn = 2.0f / lmax[0];
  for (; i < n; i += stride) {
    int node = (int)(i >> 7);
    float hv = h[i] * dsq[node];
    if (first) Xo[i] = -rn * hv + (rn - 1.0f) * Xp[i];
    else       Xo[i] = -2.0f * rn * hv + 2.0f * (rn - 1.0f) * Xp[i] - Xpp[i];
  }
}

// ---------------------------------------------------------------- eigen-filter MLP (tiny)
__global__ void eig_mlp0(const float* __restrict__ evals, const float* __restrict__ ew0,
                         const float* __restrict__ eb0, float* __restrict__ z0) {
  int i = blockIdx.x * blockDim.x + threadIdx.x;
  if (i < NEIG * HDIM) {
    int e = i >> 7, h = i & 127;
    float v = evals[e] * ew0[h] + eb0[h];
    z0[i] = v > 0.0f ? v : 0.0f;
  }
}

__global__ void eig_mlp1(const float* __restrict__ z0, const float* __restrict__ ew1,
                         const float* __restrict__ eb1, float* __restrict__ z1) {
  int i = blockIdx.x * blockDim.x + threadIdx.x;
  if (i < NEIG * HDIM) {
    int e = i >> 7, h = i & 127;
    float acc = eb1[h];
    for (int k = 0; k < HDIM; ++k) acc += z0[e * HDIM + k] * ew1[k * HDIM + h];
    z1[i] = acc > 0.0f ? acc : 0.0f;
  }
}

__global__ void eig_mlp2(const float* __restrict__ z1, const float* __restrict__ ew2,
                         const float* __restrict__ eb2, float* __restrict__ ef) {
  int e = blockIdx.x * blockDim.x + threadIdx.x;
  if (e < NEIG) {
    float acc = eb2[0];
    for (int k = 0; k < HDIM; ++k) acc += z1[e * HDIM + k] * ew2[k];
    ef[e] = acc;
  }
}

// ---------------------------------------------------------------- hs = evecs^T @ feature  (64x128)
// FP32 WMMA 16x16x4; K = node dimension, split across gridDim.z chunks of 250 steps (x4 nodes).
__global__ void hs_accum(const float* __restrict__ evecs, const float* __restrict__ feat,
                         float* __restrict__ hs) {
  int lane = threadIdx.x;
  int e0 = blockIdx.x * 16;       // eigen tile (rows of hs)
  int f0 = blockIdx.y * 16;       // feature tile (cols of hs)
  int m = lane & 15;
  int koff = (lane >> 4) * 2;     // ISA A layout: lanes 0-15 -> K=0,1 ; lanes 16-31 -> K=2,3
  v8f c = {};
  int base = blockIdx.z * 250 * 4;
  for (int s = 0; s < 250; ++s) {
    int n0 = base + s * 4;
    v2f a, b;
    a[0] = evecs[(long)(n0 + koff)     * NEIG + e0 + m];   // A[m,k] = evecs[n,e] (transposed)
    a[1] = evecs[(long)(n0 + koff + 1) * NEIG + e0 + m];
    b[0] = feat [(long)(n0 + koff)     * FDIM + f0 + m];   // B[k,f]
    b[1] = feat [(long)(n0 + koff + 1) * FDIM + f0 + m];
    c = __builtin_amdgcn_wmma_f32_16x16x4_f32(false, a, false, b, (short)0, c, false, false);
  }
  int rbase = (lane >> 4) * 8;    // D layout: VGPR r -> rows r (lanes 0-15) / r+8 (lanes 16-31)
#pragma unroll
  for (int r = 0; r < 8; ++r)
    atomicAdd(&hs[(e0 + rbase + r) * FDIM + f0 + m], c[r]);
}

__global__ void hs_scale(float* __restrict__ hs, const float* __restrict__ ef) {
  int i = blockIdx.x * blockDim.x + threadIdx.x;
  if (i < NEIG * FDIM) hs[i] *= ef[i >> 7];
}

// ---------------------------------------------------------------- Xspec = evecs @ hs  (N x 128)
__global__ void spectral_expand(const float* __restrict__ evecs, const float* __restrict__ hs,
                                float* __restrict__ out) {
  int lane = threadIdx.x;
  int n0 = blockIdx.x * 16;
  int f0 = blockIdx.y * 16;
  int m = lane & 15;
  int koff = (lane >> 4) * 2;
  v8f c = {};
#pragma unroll
  for (int k0 = 0; k0 < NEIG; k0 += 4) {
    v2f a = *(const v2f*)&evecs[(long)(n0 + m) * NEIG + k0 + koff];  // b64 load (contiguous K pair)
    v2f b;
    b[0] = hs[(k0 + koff)     * FDIM + f0 + m];
    b[1] = hs[(k0 + koff + 1) * FDIM + f0 + m];
    c = __builtin_amdgcn_wmma_f32_16x16x4_f32(false, a, false, b, (short)0, c, false, false);
  }
  int rbase = (lane >> 4) * 8;
#pragma unroll
  for (int r = 0; r < 8; ++r)
    out[(long)(n0 + rbase + r) * FDIM + f0 + m] = c[r];
}

// ---------------------------------------------------------------- fused output GEMM (TDM + LDS)
// out = [X0|X1|X2|X3|Xspec] (N,640) @ lin_w (640,128) + lin_b
// One block = 16 nodes x all 128 output features (8 waves, one 16x16 f-tile each).
// The five 16x128 A tiles are DMAd to LDS by the Tensor Data Mover with 4-dword row
// padding (132-dword LDS row stride -> conflict-free ds_load_b64). B streams from L2.
__global__ void __launch_bounds__(256) final_gemm_tdm(
    const float* __restrict__ A0, const float* __restrict__ A1,
    const float* __restrict__ A2, const float* __restrict__ A3,
    const float* __restrict__ A4,
    const float* __restrict__ W, const float* __restrict__ bias,
    float* __restrict__ out) {
  __shared__ float lds[5 * 16 * LDS_ROW];
  const int tid = threadIdx.x;
  const int n0 = blockIdx.x * 16;

  if (tid == 0) {  // TDM ignores EXEC; issue once per block, other waves skip via branch
    unsigned int lds_base = (unsigned int)(uintptr_t)&lds[0];
    const float* srcs[5] = {A0, A1, A2, A3, A4};
#pragma unroll
    for (int b = 0; b < 5; ++b) {
      unsigned long long ga = (unsigned long long)(uintptr_t)(srcs[b] + (size_t)n0 * FDIM);
      // D# group0: count=1 | lds_addr | global_addr(57b) | type=2
      u32x4 g0 = { 1u,
                   lds_base + (unsigned int)(b * 16 * LDS_ROW * 4),
                   (unsigned int)ga,
                   ((unsigned int)(ga >> 32) & 0x01FFFFFFu) | 0x80000000u };
      // D# group1: data_size=4B, pad_enable, pad_interval=128dw(code6), pad_amount=4dw(code3)
      //            tensor_dim0=128, tensor_dim1=16, tile_dim0=128, tile_dim1=16, dim0_stride=128
      i32x8 g1 = { (int)((2u << 16) | (1u << 20) | (6u << 22) | (3u << 25)),
                   (int)(128u << 16),   // tensor_dim0[15:0] @ bits 63:48
                   (int)(16u << 16),    // tensor_dim1[15:0] @ bits 95:80
                   (int)(128u << 16),   // tile_dim0 @ bits 127:112
                   16,                  // tile_dim1 @ bits 143:128
                   128,                 // tensor_dim0_stride[31:0]
                   0, 0 };
      i32x4 gz4 = {0, 0, 0, 0};
      i32x8 gz8 = {0, 0, 0, 0, 0, 0, 0, 0};
      __builtin_amdgcn_tensor_load_to_lds(g0, g1, gz4, gz4, gz8, 0);
    }
    __builtin_amdgcn_s_wait_tensorcnt(0);
  }
  __syncthreads();

  const int wave = tid >> 5;          // 8 waves -> 8 feature tiles
  const int lane = tid & 31;
  const int f0 = wave * 16;
  const int m = lane & 15;
  const int koff = (lane >> 4) * 2;
  v8f c = {};
#pragma unroll
  for (int b = 0; b < 5; ++b) {
    const float* __restrict__ Wb = W + (long)b * FDIM * 128;
    const float* lA = &lds[b * 16 * LDS_ROW + m * LDS_ROW];
    for (int k0 = 0; k0 < FDIM; k0 += 4) {
      v2f av = *(const v2f*)&lA[k0 + koff];        // ds_load_b64, conflict-free
      v2f bv;
      bv[0] = Wb[(k0 + koff)     * 128 + f0 + m];
      bv[1] = Wb[(k0 + koff + 1) * 128 + f0 + m];
      c = __builtin_amdgcn_wmma_f32_16x16x4_f32(false, av, false, bv, (short)0, c, false, false);
    }
  }
  const int rbase = (lane >> 4) * 8;
  float bb = bias[f0 + m];
#pragma unroll
  for (int r = 0; r < 8; ++r)
    out[(long)(n0 + rbase + r) * FDIM + f0 + m] = c[r] + bb;
}

// ---------------------------------------------------------------- batch norm
__global__ void col_stats(const float* __restrict__ raw, float* __restrict__ colsum,
                          float* __restrict__ colsq) {
  __shared__ float ssum[256];
  __shared__ float ssq[256];
  int col = blockIdx.x;
  float s = 0.0f, q = 0.0f;
  for (int n = threadIdx.x; n < N_NODES; n += blockDim.x) {
    float v = raw[(long)n * FDIM + col];
    s += v;
    q += v * v;
  }
  ssum[threadIdx.x] = s;
  ssq[threadIdx.x] = q;
  __syncthreads();
  for (int w = 128; w > 0; w >>= 1) {
    if (threadIdx.x < w) {
      ssum[threadIdx.x] += ssum[threadIdx.x + w];
      ssq[threadIdx.x]  += ssq[threadIdx.x + w];
    }
    __syncthreads();
  }
  if (threadIdx.x == 0) {
    colsum[col] = ssum[0];
    colsq[col]  = ssq[0];
  }
}

__global__ void bn_relu(const float* __restrict__ raw, const float* __restrict__ colsum,
                        const float* __restrict__ colsq, const float* __restrict__ gamma,
                        const float* __restrict__ beta, float* __restrict__ out) {
  long i = (long)blockIdx.x * blockDim.x + threadIdx.x;
  const long n = (long)N_NODES * FDIM;
  long stride = (long)gridDim.x * blockDim.x;
  const float invN = 1.0f / (float)N_NODES;
  for (; i < n; i += stride) {
    int col = (int)(i & 127);
    float mu = colsum[col] * invN;
    float var = colsq[col] * invN - mu * mu;
    float y = (raw[i] - mu) * rsqrtf(var + BN_EPS) * gamma[col] + beta[col];
    out[i] = y > 0.0f ? y : 0.0f;
  }
}

// ---------------------------------------------------------------- host launcher
extern "C" void kernel_launch(void* const* d_in, const int* in_sizes, int n_in,
                              void* d_out, int out_size, void* d_ws, size_t ws_size,
                              hipStream_t stream) {
  (void)in_sizes; (void)n_in; (void)out_size; (void)ws_size;
  const float*      feature = (const float*)d_in[0];
  const float*      lin_w   = (const float*)d_in[1];
  const float*      lin_b   = (const float*)d_in[2];
  const float*      ew0     = (const float*)d_in[3];
  const float*      eb0     = (const float*)d_in[4];
  const float*      ew1     = (const float*)d_in[5];
  const float*      eb1     = (const float*)d_in[6];
  const float*      ew2     = (const float*)d_in[7];
  const float*      eb2     = (const float*)d_in[8];
  const float*      gamma   = (const float*)d_in[9];
  const float*      beta    = (const float*)d_in[10];
  const float*      lmax    = (const float*)d_in[11];
  const float*      evecs   = (const float*)d_in[12];
  const float*      evals   = (const float*)d_in[13];
  const long long*  esrc    = (const long long*)d_in[14];
  const long long*  edst    = (const long long*)d_in[15];
  float* out = (float*)d_out;

  char* ws = (char*)d_ws;
  size_t off = 0;
  auto alloc = [&](size_t bytes) -> float* {
    off = (off + 255) & ~(size_t)255;
    float* p = (float*)(ws + off);
    off += bytes;
    return p;
  };
  const size_t NF_BYTES = (size_t)N_NODES * FDIM * sizeof(float);
  float* deg    = alloc((size_t)N_NODES * sizeof(float));
  float* dsq    = alloc((size_t)N_NODES * sizeof(float));
  float* X1     = alloc(NF_BYTES);
  float* X2     = alloc(NF_BYTES);
  float* X3     = alloc(NF_BYTES);
  float* bufH   = alloc(NF_BYTES);   // Laplacian accumulator; later reused as raw GEMM output
  float* Xspec  = alloc(NF_BYTES);   // spectral block (evecs @ (ef*hs))
  float* z0     = alloc((size_t)NEIG * HDIM * sizeof(float));
  float* z1     = alloc((size_t)NEIG * HDIM * sizeof(float));
  float* ef     = alloc((size_t)NEIG * sizeof(float));
  float* hs_s   = alloc((size_t)NEIG * FDIM * sizeof(float));
  float* colsum = alloc(128 * sizeof(float));
  float* colsq  = alloc(128 * sizeof(float));

  const long NF = (long)N_NODES * FDIM;

  // degree -> d_sqrt
  zero_f32<<<256, 256, 0, stream>>>(deg, N_NODES);
  deg_kernel<<<(N_EDGES + 255) / 256, 256, 0, stream>>>(edst, deg);
  dsqrt_kernel<<<(N_NODES + 255) / 256, 256, 0, stream>>>(deg, dsq);

  // eigen-filter MLP
  eig_mlp0<<<(NEIG * HDIM + 255) / 256, 256, 0, stream>>>(evals, ew0, eb0, z0);
  eig_mlp1<<<(NEIG * HDIM + 255) / 256, 256, 0, stream>>>(z0, ew1, eb1, z1);
  eig_mlp2<<<1, 64, 0, stream>>>(z1, ew2, eb2, ef);

  // spectral block: hs = evecs^T @ feature ; hs *= ef ; Xspec = evecs @ hs
  zero_f32<<<32, 256, 0, stream>>>(hs_s, NEIG * FDIM);
  hs_accum<<<dim3(NEIG / 16, FDIM / 16, 50), 32, 0, stream>>>(evecs, feature, hs_s);
  hs_scale<<<(NEIG * FDIM + 255) / 256, 256, 0, stream>>>(hs_s, ef);
  spectral_expand<<<dim3(N_NODES / 16, FDIM / 16), 32, 0, stream>>>(evecs, hs_s, Xspec);

  // Chebyshev recurrence (3 Laplacian applies)
  zero_f32<<<4096, 256, 0, stream>>>(bufH, NF);
  edge_scatter<<<N_EDGES, FDIM, 0, stream>>>(feature, dsq, esrc, edst, bufH);
  cheb_combine<<<4096, 256, 0, stream>>>(X1, bufH, dsq, feature, feature, lmax, 1);

  zero_f32<<<4096, 256, 0, stream>>>(bufH, NF);
  edge_scatter<<<N_EDGES, FDIM, 0, stream>>>(X1, dsq, esrc, edst, bufH);
  cheb_combine<<<4096, 256, 0, stream>>>(X2, bufH, dsq, X1, feature, lmax, 0);

  zero_f32<<<4096, 256, 0, stream>>>(bufH, NF);
  edge_scatter<<<N_EDGES, FDIM, 0, stream>>>(X2, dsq, esrc, edst, bufH);
  cheb_combine<<<4096, 256, 0, stream>>>(X3, bufH, dsq, X2, X1, lmax, 0);

  // fused output GEMM into bufH (reused; Laplacian done with it)
  final_gemm_tdm<<<dim3(N_NODES / 16), 256, 0, stream>>>(
      feature, X1, X2, X3, Xspec, lin_w, lin_b, bufH);

  // batchnorm + relu -> d_out (raw kept in ws so replays are idempotent)
  col_stats<<<128, 256, 0, stream>>>(bufH, colsum, colsq);
  bn_relu<<<4096, 256, 0, stream>>>(bufH, colsum, colsq, gamma, beta, out);
}